// VoxelNetwork_depth_64682207478465
// MI455X (gfx1250) — compile-verified
//
#include <hip/hip_runtime.h>
#include <hip/hip_bf16.h>
#include <stdint.h>

// depth_map_to_voxel: depth[8,256,256] -> 4x nearest upsample -> pad W by 128 ->
// pc = ray * depth (x-major flatten) -> quantize -> scatter 1.0 into [8,64,64,64].
//
// Pure scatter / data movement (~70 MB total => ~3us at 23.3 TB/s). No matmul
// content, so no WMMA; the CDNA5-specific lever used here is the async
// global->LDS DMA path (ASYNCcnt) for staging the per-block depth column.

#define VS      64
#define VS3     (VS * VS * VS)          // 262144
#define IMGH    1024
#define IMGW    1280
#define NB      8
#define DHW     256
#define NPIX    (IMGW * IMGH)           // 1310720
#define PAD_LIN ((32 * VS + 32) * VS)   // voxel hit by depth==0 points: (32,32,0)

__global__ __launch_bounds__(256) void voxel_zero_kernel(float4* __restrict__ out, int n4) {
    int i = blockIdx.x * 256 + threadIdx.x;
    if (i < n4) out[i] = make_float4(0.f, 0.f, 0.f, 0.f);
}

__global__ __launch_bounds__(256) void voxel_scatter_kernel(
    const float* __restrict__ depth,   // [8, 256, 256]
    const float* __restrict__ ray,     // [1310720, 3]
    float* __restrict__ out)           // [8, 262144]
{
    __shared__ float s_depth[NB][64];  // depth column segment for this block, all batches

    const int tid = threadIdx.x;
    const int n   = blockIdx.x * 256 + tid;  // flat pixel id, x-major: n = x*1024 + y
    const int x   = n >> 10;                 // block-uniform (256 | 1024)
    const int y0  = (blockIdx.x & 3) << 8;   // base y of this block; y = y0 + tid

    // Padded columns: depth == 0 => every pixel maps to voxel (32,32,0).
    if (x < 128 || x >= 1152) {
        if (tid < NB) out[tid * VS3 + PAD_LIN] = 1.0f;
        return;  // uniform across block: safe
    }

    const int xd  = (x - 128) >> 2;   // depth column index [0,256)
    const int yq0 = y0 >> 2;          // first depth row needed by this block

    // Async-DMA the 64 depth rows x 8 batches (2 KB) this block needs into LDS.
    // One lane per depth row; waves 0-1 (wave32) issue, all waves consume.
    if (tid < 64) {
        const unsigned lds0 =
            (unsigned)(uintptr_t)(&s_depth[0][0]) + (unsigned)(tid * 4u);
#pragma unroll
        for (int b = 0; b < NB; ++b) {
            const unsigned lds_b = lds0 + (unsigned)(b * 64 * 4);
            const unsigned long long ga = (unsigned long long)(uintptr_t)(
                depth + (size_t)b * (DHW * DHW) + (size_t)(yq0 + tid) * DHW + xd);
            asm volatile("global_load_async_to_lds_b32 %0, %1, off"
                         :: "v"(lds_b), "v"(ga)
                         : "memory");
        }
    }
    asm volatile("s_wait_asynccnt 0x0" ::: "memory");
    __syncthreads();

    // One coalesced 12B ray load per pixel, reused across all 8 batch items.
    const float3 r = *reinterpret_cast<const float3*>(ray + (size_t)n * 3);

    const float scale = (float)VS / 3.0f;  // VOLUME_SIZE / CUBOID_SIDE
    const int   yq    = tid >> 2;          // local depth row (4x nearest upsample)

#pragma unroll
    for (int b = 0; b < NB; ++b) {
        const float dv = s_depth[b][yq];
        const float px = r.x * dv;
        const float py = r.y * dv;
        const float pz = r.z * dv;
        // rintf == round-half-even (default RNE mode) == jnp.round
        const int ix = (int)rintf((px + 1.5f) * scale);
        const int iy = (int)rintf((py + 1.5f) * scale);
        const int iz = (int)rintf(pz * scale);
        if (((unsigned)ix < (unsigned)VS) &
            ((unsigned)iy < (unsigned)VS) &
            ((unsigned)iz < (unsigned)VS)) {
            // All writers store 1.0f: benign race, no atomics needed.
            out[(size_t)b * VS3 + ((ix * VS + iy) * VS + iz)] = 1.0f;
        }
    }
}

extern "C" void kernel_launch(void* const* d_in, const int* in_sizes, int n_in,
                              void* d_out, int out_size, void* d_ws, size_t ws_size,
                              hipStream_t stream) {
    (void)in_sizes; (void)n_in; (void)d_ws; (void)ws_size;
    const float* depth = (const float*)d_in[0];   // [8,256,256] f32
    const float* ray   = (const float*)d_in[1];   // [1310720,3] f32
    float*       out   = (float*)d_out;           // [8,64,64,64] f32

    // Zero the occupancy grids (harness poisons d_out).
    const int n4 = out_size / 4;                  // 524288 float4s
    voxel_zero_kernel<<<(n4 + 255) / 256, 256, 0, stream>>>((float4*)out, n4);

    // One thread per padded-image pixel; batch loop inside.
    voxel_scatter_kernel<<<NPIX / 256, 256, 0, stream>>>(depth, ray, out);
}